// TransitionPointModule_46583215292598
// MI455X (gfx1250) — compile-verified
//
#include <hip/hip_runtime.h>

typedef __attribute__((ext_vector_type(2))) float v2f;
typedef __attribute__((ext_vector_type(4))) float v4f;
typedef __attribute__((ext_vector_type(8))) float v8f;

#define N_COMP   8
#define N_BASIS  16
#define N_CONT   56
#define N_CAT    8
#define N_UNIQUE 100
#define SEPC     1e-3f
#define T1V      10.0f

#define PW_ELEMS   (N_CONT * 4 * 32 * 2)   // 14336 floats
#define PWIC_ELEMS (N_CONT * 16)           //   896 floats

__device__ __forceinline__ float softplus_f(float x) {
    // stable: max(x,0) + log(1 + exp(-|x|))
    float e = __expf(-fabsf(x));
    return fmaxf(x, 0.0f) + __logf(1.0f + e);
}

// ---------------------------------------------------------------------------
// Prepass: repack Wh(16,56,8)|Wv(16,56,8) into WMMA B-operand lane layout:
//   pW[((i*4 + j)*32 + L)*2 + e] = W[k = 4j + 2*(L>>4) + e][i][n = L&15]
//   (n<8 -> Wh channel n ; n>=8 -> Wv channel n-8)
// and Wic(16,56,1) into per-k-group broadcast layout:
//   pWic[(i*2 + g)*8 + 2j + e] = Wic[4j + 2g + e][i]
// ---------------------------------------------------------------------------
__global__ void repack_weights(const float* __restrict__ Wh,
                               const float* __restrict__ Wv,
                               const float* __restrict__ Wic,
                               float* __restrict__ pW,
                               float* __restrict__ pWic) {
    int q = blockIdx.x * blockDim.x + threadIdx.x;
    if (q < PW_ELEMS) {
        int e  = q & 1;
        int L  = (q >> 1) & 31;
        int j  = (q >> 6) & 3;
        int i  = q >> 8;
        int n  = L & 15;
        int hi = L >> 4;
        int k  = 4 * j + 2 * hi + e;
        float v = (n < 8) ? Wh[(k * N_CONT + i) * N_COMP + n]
                          : Wv[(k * N_CONT + i) * N_COMP + (n - 8)];
        pW[q] = v;
    } else {
        int q2 = q - PW_ELEMS;
        if (q2 < PWIC_ELEMS) {
            int e  = q2 & 1;
            int j  = (q2 >> 1) & 3;
            int g  = (q2 >> 3) & 1;
            int i  = q2 >> 4;
            pWic[q2] = Wic[(4 * j + 2 * g + e) * N_CONT + i];
        }
    }
}

// ---------------------------------------------------------------------------
// Main kernel: 1 wave (32 lanes) handles a tile of 16 samples.
// Block = 256 threads = 8 waves = 128 samples.
// ---------------------------------------------------------------------------
__global__ __launch_bounds__(256) void tpm_kernel(
    const float* __restrict__ B,        // (batch, 56, 16)
    const float* __restrict__ B_cat,    // (batch, 800) one-hot
    const float* __restrict__ Wh_cat,   // (8, 100, 8)
    const float* __restrict__ bh,       // (8,)
    const float* __restrict__ Wv_cat,   // (8, 100, 8)
    const float* __restrict__ bv,       // (8,)
    const float* __restrict__ Wic_cat,  // (8, 100, 1)
    const float* __restrict__ bic,      // (1,)
    const float* __restrict__ pW,       // repacked Wh|Wv
    const float* __restrict__ pWic,     // repacked Wic
    float* __restrict__ out)            // (batch, 9, 2)
{
    __shared__ float hv[8][16 * 17];       // per-wave: [sample m][ch 0..15, 16=IC]
    __shared__ int   codes_lds[8][16][8];  // per-wave: [sample m][cat feature]

    const int tid  = threadIdx.x;
    const int w    = tid >> 5;
    const int lane = tid & 31;
    const int m16  = lane & 15;   // A-row sample index / D-column channel index
    const int hi   = lane >> 4;
    const long b0  = ((long)blockIdx.x * 8 + w) * 16;

    if (lane < 16) hv[w][lane * 17 + 16] = 0.0f;   // zero IC column
    __syncthreads();

    // ---------------- Phase 1: WMMA over 56 continuous features --------------
    float Sacc[8];
#pragma unroll
    for (int r = 0; r < 8; ++r) Sacc[r] = 0.0f;
    float icacc = 0.0f;

    const float* bp      = B + (size_t)(b0 + m16) * (N_CONT * N_BASIS) + 2 * hi;
    const v2f*   pW2     = (const v2f*)pW;
    const v4f*   pWic4   = (const v4f*)pWic;

    for (int i = 0; i < N_CONT; ++i) {
        const float* bpi = bp + i * N_BASIS;
        v2f a0 = *(const v2f*)(bpi + 0);    // A: k = 4j + 2*hi + {0,1}
        v2f a1 = *(const v2f*)(bpi + 4);
        v2f a2 = *(const v2f*)(bpi + 8);
        v2f a3 = *(const v2f*)(bpi + 12);
        v2f w0 = pW2[(i * 4 + 0) * 32 + lane];
        v2f w1 = pW2[(i * 4 + 1) * 32 + lane];
        v2f w2 = pW2[(i * 4 + 2) * 32 + lane];
        v2f w3 = pW2[(i * 4 + 3) * 32 + lane];
        v4f wic_lo = pWic4[(i * 2 + hi) * 2 + 0];
        v4f wic_hi = pWic4[(i * 2 + hi) * 2 + 1];

        v8f d = {};
        d = __builtin_amdgcn_wmma_f32_16x16x4_f32(false, a0, false, w0, (short)0, d, false, false);
        d = __builtin_amdgcn_wmma_f32_16x16x4_f32(false, a1, false, w1, (short)0, d, false, false);
        d = __builtin_amdgcn_wmma_f32_16x16x4_f32(false, a2, false, w2, (short)0, d, false, false);
        d = __builtin_amdgcn_wmma_f32_16x16x4_f32(false, a3, false, w3, (short)0, d, false, false);

        // IC channel: plain dot (no per-feature nonlinearity -> one long dot)
        icacc += a0.x * wic_lo.x + a0.y * wic_lo.y
               + a1.x * wic_lo.z + a1.y * wic_lo.w
               + a2.x * wic_hi.x + a2.y * wic_hi.y
               + a3.x * wic_hi.z + a3.y * wic_hi.w;

        // per-feature softplus, then accumulate across features
#pragma unroll
        for (int r = 0; r < 8; ++r) Sacc[r] += softplus_f(d[r]);
    }

    // D layout: VGPR r -> (M = r + 8*hi, N = lane&15). Transpose into LDS.
#pragma unroll
    for (int r = 0; r < 8; ++r) hv[w][(r + hi * 8) * 17 + m16] = Sacc[r];
    atomicAdd(&hv[w][m16 * 17 + 16], icacc);   // halves of sample m16's IC dot

    // ---------------- Phase 2a: recover one-hot codes ------------------------
    {
        const long b = b0 + m16;
        for (int f = hi * 4; f < hi * 4 + 4; ++f) {
            const v4f* cp = (const v4f*)(B_cat + (size_t)b * (N_CAT * N_UNIQUE)
                                         + f * N_UNIQUE);
            int idx = 0;
#pragma unroll 5
            for (int t = 0; t < 25; ++t) {
                v4f v = cp[t];
                idx = (v.x > 0.5f) ? 4 * t + 0 : idx;
                idx = (v.y > 0.5f) ? 4 * t + 1 : idx;
                idx = (v.z > 0.5f) ? 4 * t + 2 : idx;
                idx = (v.w > 0.5f) ? 4 * t + 3 : idx;
            }
            codes_lds[w][m16][f] = idx;
        }
    }
    __syncthreads();

    // ---------------- Phase 2b: categorical contributions (conflict-free) ----
    {
        const int m = m16;
        const float* wcat = hi ? Wv_cat : Wh_cat;
        const int colbase = hi ? 8 : 0;
        float loc[8];
#pragma unroll
        for (int c = 0; c < 8; ++c) loc[c] = 0.0f;
        float icl = 0.0f;
        for (int f = 0; f < N_CAT; ++f) {
            int idx = codes_lds[w][m][f];
            const float* wp = wcat + ((size_t)f * N_UNIQUE + idx) * N_COMP;
#pragma unroll
            for (int c = 0; c < 8; ++c) loc[c] += softplus_f(wp[c]);
            if (hi == 0) icl += Wic_cat[f * N_UNIQUE + idx];
        }
#pragma unroll
        for (int c = 0; c < 8; ++c) hv[w][m * 17 + colbase + c] += loc[c];
        if (hi == 0) hv[w][m * 17 + 16] += icl;
    }
    __syncthreads();

    // ---------------- Phase 3: cumsums + output (lanes 0..15, 1 sample each) --
    if (lane < 16) {
        const int m = lane;
        const long b = b0 + m;
        float hs[8], vs[8];
#pragma unroll
        for (int c = 0; c < 8; ++c) {
            hs[c] = hv[w][m * 17 + c]     + softplus_f(bh[c]);
            vs[c] = hv[w][m * 17 + 8 + c] + softplus_f(bv[c]);
        }
        float x0 = hv[w][m * 17 + 16] + bic[0];

        float tarr[9], xarr[9];
        tarr[0] = 0.0f;
        float acc = 0.0f;
#pragma unroll
        for (int c = 0; c < 8; ++c) {
            acc += hs[c];
            tarr[c + 1] = acc + SEPC * (float)(c + 1);
        }
        tarr[8] = fmaxf(tarr[7] + SEPC, T1V);

        xarr[0] = x0;
        float xacc = x0;
#pragma unroll
        for (int c = 0; c < 8; ++c) {
            xacc += ((c & 1) ? -1.0f : 1.0f) * vs[c];
            xarr[c + 1] = xacc;
        }

        v2f* op = (v2f*)(out + (size_t)b * 18);
#pragma unroll
        for (int k = 0; k < 9; ++k) {
            v2f o; o.x = tarr[k]; o.y = xarr[k];
            op[k] = o;
        }
    }
}

// ---------------------------------------------------------------------------
extern "C" void kernel_launch(void* const* d_in, const int* in_sizes, int n_in,
                              void* d_out, int out_size, void* d_ws, size_t ws_size,
                              hipStream_t stream) {
    const float* B       = (const float*)d_in[0];
    const float* B_cat   = (const float*)d_in[1];
    const float* Wh      = (const float*)d_in[2];
    const float* Wh_cat  = (const float*)d_in[3];
    const float* bh      = (const float*)d_in[4];
    const float* Wv      = (const float*)d_in[5];
    const float* Wv_cat  = (const float*)d_in[6];
    const float* bv      = (const float*)d_in[7];
    const float* Wic     = (const float*)d_in[8];
    const float* Wic_cat = (const float*)d_in[9];
    const float* bic     = (const float*)d_in[10];

    float* pW   = (float*)d_ws;
    float* pWic = pW + PW_ELEMS;

    const int batch  = in_sizes[0] / (N_CONT * N_BASIS);   // 32768
    const int repackThreads = PW_ELEMS + PWIC_ELEMS;       // 15232

    repack_weights<<<(repackThreads + 255) / 256, 256, 0, stream>>>(
        Wh, Wv, Wic, pW, pWic);

    const int tiles  = batch / 16;   // 2048 waves
    const int blocks = tiles / 8;    // 256 blocks of 8 waves
    tpm_kernel<<<blocks, 256, 0, stream>>>(
        B, B_cat, Wh_cat, bh, Wv_cat, bv, Wic_cat, bic, pW, pWic, (float*)d_out);
}